// MeanAggregator_14955076125085
// MI455X (gfx1250) — compile-verified
//
#include <hip/hip_runtime.h>
#include <stdint.h>

#define D 128
#define TILE_EDGES 64
#define TILE_ELEMS (TILE_EDGES * D)          // 8192 floats = 32 KB per buffer
#define BLOCK 256                            // 8 waves (wave32)
#define WAVES (BLOCK / 32)
#define EDGES_PER_WAVE (TILE_EDGES / WAVES)  // 8
#define GRID 1024

typedef float v4f __attribute__((ext_vector_type(4)));
typedef unsigned int u32x4 __attribute__((ext_vector_type(4)));
typedef int i32x4 __attribute__((ext_vector_type(4)));
typedef int i32x8 __attribute__((ext_vector_type(8)));

// Build a TDM descriptor for a contiguous 1-D copy of TILE_ELEMS fp32 from
// global memory into LDS, clamped to `elems_remaining` valid elements
// (OOB tensor reads return zero per CDNA5 ISA).
__device__ __forceinline__ void tdm_load_tile(uint32_t lds_off,
                                              const float* gbase,
                                              long long elems_remaining) {
  uint64_t ga = (uint64_t)(uintptr_t)gbase;
  uint32_t nvalid =
      (elems_remaining <= 0) ? 0u
      : (elems_remaining >= 0x7FFFFFFFll ? 0x7FFFFFFFu
                                         : (uint32_t)elems_remaining);
  u32x4 g0;
  g0.x = 1u;                                    // count=1 (valid descriptor)
  g0.y = lds_off;                               // lds_addr (bytes)
  g0.z = (uint32_t)(ga & 0xFFFFFFFFull);        // global_addr[31:0]
  g0.w = (uint32_t)((ga >> 32) & 0x01FFFFFFull) // global_addr[56:32]
         | (2u << 30);                          // type = 2 ("image")
  i32x8 g1;
  g1[0] = (int)(2u << 16);                      // data_size = 2 (4 bytes)
  g1[1] = (int)((nvalid & 0xFFFFu) << 16);      // tensor_dim0[15:0]
  g1[2] = (int)((nvalid >> 16) | (1u << 16));   // tensor_dim0[31:16] | tensor_dim1=1
  g1[3] = (int)((uint32_t)TILE_ELEMS << 16);    // tile_dim0 = 8192
  g1[4] = 1;                                    // tile_dim1 = 1, tile_dim2 = 0
  g1[5] = TILE_ELEMS;                           // tensor_dim0_stride (lo 32)
  g1[6] = 0;
  g1[7] = 0;
  i32x4 gz4 = {0, 0, 0, 0};                     // groups 2/3 unused (<=2D)
  i32x8 gz8 = {0, 0, 0, 0, 0, 0, 0, 0};        // extra group (clang-23 form)
  __builtin_amdgcn_tensor_load_to_lds(g0, g1, gz4, gz4, gz8, /*cpol*/ 0);
}

__global__ void zero_kernel(float* __restrict__ out,
                            unsigned int* __restrict__ cnt,
                            long long tot_out, int n) {
  long long i = (long long)blockIdx.x * blockDim.x + threadIdx.x;
  long long stride = (long long)gridDim.x * blockDim.x;
  for (long long j = i; j < tot_out; j += stride) out[j] = 0.0f;
  for (long long j = i; j < n; j += stride) cnt[j] = 0u;
}

__global__ __launch_bounds__(BLOCK) void scatter_kernel(
    const float* __restrict__ msg, const long long* __restrict__ index,
    float* __restrict__ out, unsigned int* __restrict__ cnt,
    long long E, int tiles_total, int tiles_per_block) {
  __shared__ float smem[2][TILE_ELEMS];  // 64 KB double buffer

  const int tid = threadIdx.x;
  const int wave = tid >> 5;
  const int lane = tid & 31;

  const int tile0 = blockIdx.x * tiles_per_block;
  const int tile_end = min(tile0 + tiles_per_block, tiles_total);
  const int nt = tile_end - tile0;
  if (nt <= 0) return;  // uniform per block: safe w.r.t. barriers

  const uint32_t lds0 = (uint32_t)(uintptr_t)&smem[0][0];
  const uint32_t lds1 = (uint32_t)(uintptr_t)&smem[1][0];
  const long long total_elems = E * (long long)D;

  // Prime the pipeline: wave 0 issues the TDM load for tile 0.
  if (wave == 0) {
    long long es = (long long)tile0 * TILE_EDGES * D;
    tdm_load_tile(lds0, msg + es, total_elems - es);
  }

  for (int tt = 0; tt < nt; ++tt) {
    if (wave == 0) {
      if (tt + 1 < nt) {
        long long es = (long long)(tile0 + tt + 1) * TILE_EDGES * D;
        tdm_load_tile(((tt + 1) & 1) ? lds1 : lds0, msg + es,
                      total_elems - es);
        // TDM ops complete in order: cnt<=1 means tile tt has landed.
        __builtin_amdgcn_s_wait_tensorcnt(1);
      } else {
        __builtin_amdgcn_s_wait_tensorcnt(0);
      }
    }
    __syncthreads();

    const float* buf = smem[tt & 1];
    const long long tile_edge0 = (long long)(tile0 + tt) * TILE_EDGES;
#pragma unroll
    for (int k = 0; k < EDGES_PER_WAVE; ++k) {
      const int r = wave * EDGES_PER_WAVE + k;
      const long long e = tile_edge0 + r;
      if (e < E) {
        const int seg = (int)index[e];
        v4f v = *(const v4f*)(buf + r * D + lane * 4);
        float* o = out + (size_t)seg * D + lane * 4;
        __hip_atomic_fetch_add(o + 0, v.x, __ATOMIC_RELAXED,
                               __HIP_MEMORY_SCOPE_AGENT);
        __hip_atomic_fetch_add(o + 1, v.y, __ATOMIC_RELAXED,
                               __HIP_MEMORY_SCOPE_AGENT);
        __hip_atomic_fetch_add(o + 2, v.z, __ATOMIC_RELAXED,
                               __HIP_MEMORY_SCOPE_AGENT);
        __hip_atomic_fetch_add(o + 3, v.w, __ATOMIC_RELAXED,
                               __HIP_MEMORY_SCOPE_AGENT);
        if (lane == 0)
          __hip_atomic_fetch_add(&cnt[seg], 1u, __ATOMIC_RELAXED,
                                 __HIP_MEMORY_SCOPE_AGENT);
      }
    }
    __syncthreads();  // buffer tt&1 may be overwritten next iteration
  }
}

__global__ void div_kernel(float* __restrict__ out,
                           const unsigned int* __restrict__ cnt,
                           long long tot_out) {
  long long i = (long long)blockIdx.x * blockDim.x + threadIdx.x;
  long long stride = (long long)gridDim.x * blockDim.x;
  for (long long j = i; j < tot_out; j += stride) {
    float denom = fmaxf((float)cnt[j >> 7], 1.0f);
    out[j] = out[j] / denom;
  }
}

extern "C" void kernel_launch(void* const* d_in, const int* in_sizes, int n_in,
                              void* d_out, int out_size, void* d_ws,
                              size_t ws_size, hipStream_t stream) {
  const float* msg = (const float*)d_in[0];
  const long long* index = (const long long*)d_in[1];
  // d_in[2] (t) is unused by the reference.
  const long long E = (long long)in_sizes[0] / D;  // 2,000,000
  const int N = out_size / D;                      // 100,000
  float* out = (float*)d_out;
  unsigned int* cnt = (unsigned int*)d_ws;         // N u32 counters
  const long long tot_out = (long long)N * D;

  zero_kernel<<<2048, 256, 0, stream>>>(out, cnt, tot_out, N);

  const int tiles_total = (int)((E + TILE_EDGES - 1) / TILE_EDGES);
  const int tiles_per_block = (tiles_total + GRID - 1) / GRID;
  scatter_kernel<<<GRID, BLOCK, 0, stream>>>(msg, index, out, cnt, E,
                                             tiles_total, tiles_per_block);

  div_kernel<<<2048, 256, 0, stream>>>(out, cnt, tot_out);
}